// SO3SelfInteraction_17248588661209
// MI455X (gfx1250) — compile-verified
//
#include <hip/hip_runtime.h>

// ---------------------------------------------------------------------------
// SO(3) self-interaction (CG tensor product), LMAX=2: S=9 spherical comps,
// 128 features, 32000 atoms.  y[a,k,f] = sum_u C[k,u] * x[a,i_u,f]*x[a,j_u,f]
// over the 45 unordered component pairs; C (16x48 padded) is built on device
// from weight/cg/idx inputs by a prep kernel, stored in d_ws in the exact
// per-lane A-fragment layout of V_WMMA_F32_16X16X4_F32.
// Memory-bound: ~295 MB traffic -> ~12.7us @ 23.3 TB/s; math on f32 WMMA.
// ---------------------------------------------------------------------------

typedef float v2f __attribute__((ext_vector_type(2)));
typedef float v8f __attribute__((ext_vector_type(8)));

#define SDIM   9      // (LMAX+1)^2
#define NFEAT  128
#define NPAIR  45     // unordered pairs (i<=j) over 9 components
#define NCOL   48     // padded to multiple of 4 (WMMA K=4 chunks)
#define NCHUNK 12     // NCOL / 4

// Compile-time pair enumeration: column u <-> (a[u] <= b[u]).
struct PairTab { int a[NCOL]; int b[NCOL]; };
static constexpr PairTab makePairs() {
  PairTab t{};
  int u = 0;
  for (int i = 0; i < SDIM; ++i)
    for (int j = i; j < SDIM; ++j) { t.a[u] = i; t.b[u] = j; ++u; }
  for (; u < NCOL; ++u) { t.a[u] = 0; t.b[u] = 0; }
  return t;
}
static constexpr PairTab PT = makePairs();

// ---------------------------------------------------------------------------
// Prep: fold weight[widx1,widx2]*cg over paths onto the 45 unordered pairs,
// emit dense 16x48 C directly in A-fragment order:
//   afrag[(c*32 + lane)*2 + r] = C[M][4c + 2*(lane>>4) + r],  M = lane & 15
// (ISA 7.12.2, 32-bit A 16x4: lanes 0-15 hold K=0,1; lanes 16-31 hold K=2,3)
// ---------------------------------------------------------------------------
__global__ void so3_prep_kernel(const float* __restrict__ weight,
                                const float* __restrict__ cg,
                                const int* __restrict__ i1,
                                const int* __restrict__ i2,
                                const int* __restrict__ io,
                                const int* __restrict__ w1,
                                const int* __restrict__ w2,
                                int nPaths,
                                float* __restrict__ afrag) {
  int e = blockIdx.x * blockDim.x + threadIdx.x;
  if (e >= NCHUNK * 64) return;
  int c    = e >> 6;
  int rem  = e & 63;
  int lane = rem >> 1;
  int r    = rem & 1;
  int h    = lane >> 4;
  int M    = lane & 15;
  int q    = 4 * c + 2 * h + r;   // column of C this element holds
  float v = 0.0f;
  if (M < SDIM && q < NPAIR) {
    for (int p = 0; p < nPaths; ++p) {
      if (io[p] != M) continue;
      int ia = i1[p], ib = i2[p];
      int lo = ia < ib ? ia : ib;
      int hi = ia < ib ? ib : ia;
      int u  = 9 * lo - (lo * (lo - 1)) / 2 + (hi - lo);  // unordered pair id
      if (u != q) continue;
      v += weight[w1[p] * 3 + w2[p]] * cg[p];
    }
  }
  afrag[(c * 32 + lane) * 2 + r] = v;
}

// ---------------------------------------------------------------------------
// Main: 1 block (256 thr = 8 waves) per atom; wave w handles features
// [16w, 16w+16).  D(16x16) = C(16x4-chunks) x z(4x16-chunks) via 12 chained
// v_wmma_f32_16x16x4_f32.  B-fragment (ISA-mirrored layout assumption):
// lane<16 holds N=lane, K=0,1; lane>=16 holds N=lane-16, K=2,3.
// D layout: VGPR r = row M=r (lanes 0-15) / M=r+8 (lanes 16-31).
// ---------------------------------------------------------------------------
__global__ void __launch_bounds__(256)
so3_main_kernel(const float* __restrict__ x,
                const float* __restrict__ afrag,
                float* __restrict__ y) {
  const int atom = blockIdx.x;
  const int wave = threadIdx.x >> 5;          // 0..7 -> feature tile
  const int lane = threadIdx.x & 31;
  const int fcol = lane & 15;
  const int f    = wave * 16 + fcol;
  const bool hi  = (lane >= 16);

  const float* xa = x + (size_t)atom * (SDIM * NFEAT);

  // 9 streaming f32 loads per lane (lane halves share cachelines -> 1 fill).
  float xs[SDIM];
#pragma unroll
  for (int s = 0; s < SDIM; ++s)
    xs[s] = __builtin_nontemporal_load(&xa[s * NFEAT + f]);

  // All 45 pair products once per lane (pure register math).
  float z[NCOL];
#pragma unroll
  for (int u = 0; u < NPAIR; ++u) z[u] = xs[PT.a[u]] * xs[PT.b[u]];
#pragma unroll
  for (int u = NPAIR; u < NCOL; ++u) z[u] = 0.0f;

  v8f acc = {};
  const v2f* aw = (const v2f*)afrag;          // 3 KB, L0/L2 resident
#pragma unroll
  for (int c = 0; c < NCHUNK; ++c) {
    v2f a = aw[c * 32 + lane];                // A fragment (constant matrix C)
    v2f b;                                    // B fragment: per-lane-half K split
    b.x = hi ? z[4 * c + 2] : z[4 * c + 0];
    b.y = hi ? z[4 * c + 3] : z[4 * c + 1];
    // D = A(16x4) * B(4x16) + C   -- f32 WMMA, full precision
    acc = __builtin_amdgcn_wmma_f32_16x16x4_f32(
        /*neg_a=*/false, a, /*neg_b=*/false, b,
        /*c_mod=*/(short)0, acc, /*reuse_a=*/false, /*reuse_b=*/false);
  }

  // Store valid rows 0..8.  VGPR0: lanes 0-15 -> row 0, lanes 16-31 -> row 8.
  float* ya = y + (size_t)atom * (SDIM * NFEAT);
  __builtin_nontemporal_store(acc[0], &ya[(hi ? 8 : 0) * NFEAT + f]);
  if (!hi) {
#pragma unroll
    for (int r = 1; r < 8; ++r)
      __builtin_nontemporal_store(acc[r], &ya[r * NFEAT + f]);
  }
}

// ---------------------------------------------------------------------------
extern "C" void kernel_launch(void* const* d_in, const int* in_sizes, int n_in,
                              void* d_out, int out_size, void* d_ws, size_t ws_size,
                              hipStream_t stream) {
  const float* x      = (const float*)d_in[0];
  const float* weight = (const float*)d_in[1];
  const float* cg     = (const float*)d_in[2];
  const int*   idx1   = (const int*)d_in[3];
  const int*   idx2   = (const int*)d_in[4];
  const int*   idxo   = (const int*)d_in[5];
  const int*   widx1  = (const int*)d_in[6];
  const int*   widx2  = (const int*)d_in[7];
  float*       y      = (float*)d_out;
  float*       afrag  = (float*)d_ws;        // NCHUNK*64 floats = 3 KB

  const int nPaths = in_sizes[2];
  const int nAtoms = in_sizes[0] / (SDIM * NFEAT);

  so3_prep_kernel<<<(NCHUNK * 64 + 255) / 256, 256, 0, stream>>>(
      weight, cg, idx1, idx2, idxo, widx1, widx2, nPaths, afrag);

  so3_main_kernel<<<nAtoms, 256, 0, stream>>>(x, afrag, y);
}